// SoftDTW_69157563400781
// MI455X (gfx1250) — compile-verified
//
#include <hip/hip_runtime.h>
#include <hip/hip_bf16.h>

#define LSEQ 1024
#define CDIM 64
#define BATCH 64
#define BIGF 10000000000.0f
#define GAMMA_F 1.0f

typedef float v2f __attribute__((ext_vector_type(2)));
typedef float v8f __attribute__((ext_vector_type(8)));

// -------------------------------------------------------------------------
// Row squared-norms: one thread per (b, row), row is 64 contiguous floats.
// -------------------------------------------------------------------------
__global__ __launch_bounds__(256) void row_norms_kernel(
    const float* __restrict__ src, float* __restrict__ dst)
{
    const int row = blockIdx.x * blockDim.x + threadIdx.x;   // 0 .. B*L-1
    const float* p = src + (size_t)row * CDIM;
    float s = 0.0f;
#pragma unroll
    for (int c = 0; c < CDIM; c += 4) {
        float4 v = *(const float4*)(p + c);
        s += v.x * v.x + v.y * v.y + v.z * v.z + v.w * v.w;
    }
    dst[row] = s;
}

// -------------------------------------------------------------------------
// Packed diagonal-major offset for an L x L matrix.
// Diagonal d0 = i + j (0-based), r = i - max(0, d0 - (L-1)).
// -------------------------------------------------------------------------
__device__ __forceinline__ int diag_off(int d0)
{
    return (d0 < LSEQ) ? ((d0 * (d0 + 1)) >> 1)
                       : (LSEQ * LSEQ - (((2 * LSEQ - 1 - d0) * (2 * LSEQ - d0)) >> 1));
}

// -------------------------------------------------------------------------
// Batched distance GEMM: D = (x2 + y2 - 2 x.y^T) / gamma, stored
// diagonal-packed. 64x64 tile per workgroup, 8 waves x 2 WMMA tiles,
// f32 WMMA 16x16x4, LDS-staged operands.
// -------------------------------------------------------------------------
__global__ __launch_bounds__(256) void dist_gemm_kernel(
    const float* __restrict__ x, const float* __restrict__ y,
    const float* __restrict__ xn, const float* __restrict__ yn,
    float* __restrict__ Dws)
{
    __shared__ float As[64][68];   // 68-float row stride: 16B aligned, conflict-free column reads
    __shared__ float Bs[64][68];

    const int b  = blockIdx.z;
    const int m0 = blockIdx.y * 64;
    const int n0 = blockIdx.x * 64;
    const int t  = threadIdx.x;

    const float* xb = x + (size_t)b * LSEQ * CDIM;
    const float* yb = y + (size_t)b * LSEQ * CDIM;

    // Cooperative coalesced load of the 64x64 operand tiles (float4 per lane).
#pragma unroll
    for (int p = 0; p < 4; ++p) {
        int idx = t + p * 256;          // 0..1023
        int row = idx >> 4;             // 0..63
        int c4  = (idx & 15) << 2;      // 0,4,...,60
        float4 xa = *(const float4*)(xb + (size_t)(m0 + row) * CDIM + c4);
        float4 ya = *(const float4*)(yb + (size_t)(n0 + row) * CDIM + c4);
        *(float4*)&As[row][c4] = xa;
        *(float4*)&Bs[row][c4] = ya;
    }
    __syncthreads();

    const int w    = t >> 5;            // wave 0..7
    const int lane = t & 31;
    const int half = lane >> 4;         // 0/1
    const int lm   = lane & 15;
    const int msub = w & 3;             // M subtile 0..3
    const int nb   = (w >> 2) << 1;     // N subtile pair base: 0 or 2

    const int ar  = msub * 16 + lm;
    const int br0 = nb * 16 + lm;
    const int br1 = (nb + 1) * 16 + lm;

    v8f acc0 = {};
    v8f acc1 = {};

    // A fragment layout (ISA 32-bit 16x4 A): lane%16 = M, K = vgpr + 2*(lane/16).
    // B assumed symmetric (lane%16 = N). Fragments are contiguous K pairs -> b64 LDS loads.
#pragma unroll
    for (int kk = 0; kk < CDIM; kk += 4) {
        v2f a  = *(const v2f*)&As[ar][kk + 2 * half];
        v2f b0 = *(const v2f*)&Bs[br0][kk + 2 * half];
        v2f b1 = *(const v2f*)&Bs[br1][kk + 2 * half];
        acc0 = __builtin_amdgcn_wmma_f32_16x16x4_f32(
            false, a, false, b0, (short)0, acc0, false, false);
        acc1 = __builtin_amdgcn_wmma_f32_16x16x4_f32(
            false, a, false, b1, (short)0, acc1, false, false);
    }

    // Epilogue: add norms, scale, scatter into diagonal-packed layout.
    float* Db = Dws + (size_t)b * LSEQ * LSEQ;
    const float inv_gamma = 1.0f / GAMMA_F;
    const int N0 = n0 + nb * 16 + lm;
    const int N1 = n0 + (nb + 1) * 16 + lm;
    const float yn0 = yn[b * LSEQ + N0];
    const float yn1 = yn[b * LSEQ + N1];

#pragma unroll
    for (int k = 0; k < 8; ++k) {
        const int   M   = m0 + msub * 16 + k + 8 * half;
        const float xnv = xn[b * LSEQ + M];
        {
            float v  = (xnv + yn0 - 2.0f * acc0[k]) * inv_gamma;
            int   d0 = M + N0;
            int   r  = M - ((d0 > LSEQ - 1) ? (d0 - (LSEQ - 1)) : 0);
            Db[diag_off(d0) + r] = v;
        }
        {
            float v  = (xnv + yn1 - 2.0f * acc1[k]) * inv_gamma;
            int   d0 = M + N1;
            int   r  = M - ((d0 > LSEQ - 1) ? (d0 - (LSEQ - 1)) : 0);
            Db[diag_off(d0) + r] = v;
        }
    }
}

// -------------------------------------------------------------------------
// Anti-diagonal soft-DTW DP. One 1024-thread workgroup per batch.
// Three rotating diagonal buffers in LDS; coalesced diagonal-packed D reads;
// prefetch of the next diagonal; one barrier per step.
// -------------------------------------------------------------------------
__global__ __launch_bounds__(1024) void softdtw_dp_kernel(
    const float* __restrict__ Dws, float* __restrict__ out)
{
    __shared__ float rb[3][1056];    // buffers for diagonals (index = diag % 3), i = 0..1024

    const int b = blockIdx.x;
    const int t = threadIdx.x;
    const int i = t + 1;             // row index 1..1024
    const float* Db = Dws + (size_t)b * LSEQ * LSEQ;

    // diag 0: R[0,0] = 0, rest BIG.  diag 1: all BIG.
    rb[0][i] = BIGF;
    rb[1][i] = BIGF;
    if (t == 0) { rb[0][0] = 0.0f; rb[1][0] = BIGF; rb[2][0] = BIGF; }
    __syncthreads();

    float result = 0.0f;

    for (int d = 2; d <= 2 * LSEQ; ++d) {
        const float* p2 = rb[(d - 2) % 3];   // diagonal d-2
        const float* p1 = rb[(d - 1) % 3];   // diagonal d-1
        float*       cu = rb[d % 3];         // diagonal d (being written)

        const int  j   = d - i;
        const bool inR = (j >= 1) && (j <= LSEQ);

        float dval = 0.0f;
        if (inR) {
            const int d0 = d - 2;
            const int r  = (i - 1) - ((d0 > LSEQ - 1) ? (d0 - (LSEQ - 1)) : 0);
            dval = Db[diag_off(d0) + r];
            // Prefetch next diagonal's element for this row.
            if (d < 2 * LSEQ) {
                const int d1 = d - 1;
                const int r1 = (i - 1) - ((d1 > LSEQ - 1) ? (d1 - (LSEQ - 1)) : 0);
                if (r1 >= 0)
                    __builtin_prefetch(&Db[diag_off(d1) + r1], 0, 1);
            }
        }

        const float a   = p2[i - 1];   // R[i-1, j-1]
        const float u   = p1[i - 1];   // R[i-1, j]
        const float lft = p1[i];       // R[i, j-1]

        const float m  = fminf(a, fminf(u, lft));
        const float s  = __expf(m - a) + __expf(m - u) + __expf(m - lft);
        const float sm = m - __logf(s);

        const float nv = inR ? (dval + sm) : BIGF;
        cu[i] = nv;
        if (t == 0) cu[0] = BIGF;      // R[0, d] = BIG for d >= 1

        if (d == 2 * LSEQ && i == LSEQ) result = nv;
        __syncthreads();
    }

    if (i == LSEQ) out[b] = result * GAMMA_F;
}

// -------------------------------------------------------------------------
// Launch
// -------------------------------------------------------------------------
extern "C" void kernel_launch(void* const* d_in, const int* in_sizes, int n_in,
                              void* d_out, int out_size, void* d_ws, size_t ws_size,
                              hipStream_t stream)
{
    const float* x = (const float*)d_in[0];
    const float* y = (const float*)d_in[1];
    float* out = (float*)d_out;

    // Workspace layout: [ D packed-diagonal : B*L*L floats ][ xn : B*L ][ yn : B*L ]
    float* Dws = (float*)d_ws;
    float* xn  = Dws + (size_t)BATCH * LSEQ * LSEQ;
    float* yn  = xn + (size_t)BATCH * LSEQ;

    row_norms_kernel<<<(BATCH * LSEQ) / 256, 256, 0, stream>>>(x, xn);
    row_norms_kernel<<<(BATCH * LSEQ) / 256, 256, 0, stream>>>(y, yn);

    dim3 gemm_grid(LSEQ / 64, LSEQ / 64, BATCH);   // (16, 16, 64)
    dist_gemm_kernel<<<gemm_grid, 256, 0, stream>>>(x, y, xn, yn, Dws);

    softdtw_dp_kernel<<<BATCH, 1024, 0, stream>>>(Dws, out);
}